// GNNAutoencoder_3736621548267
// MI455X (gfx1250) — compile-verified
//
#include <hip/hip_runtime.h>
#include <cstdint>
#include <cstddef>

// ---------------------------------------------------------------------------
// GNN autoencoder for MI455X (gfx1250, wave32).
//  - Dense GEMMs use V_WMMA_F32_16X16X4_F32 (native fp32 WMMA, matches the
//    fp32 reference). One wave computes a 16x64 output tile (4 accumulators).
//  - Edge aggregation (the bandwidth-dominant phase, ~L2 resident) uses
//    16-lanes-per-edge float4 gathers + f32 global atomics.
//  - Softmax segment-max uses an order-preserving uint key + atomicMax(u32).
// ---------------------------------------------------------------------------

typedef float v2f __attribute__((ext_vector_type(2)));
typedef float v8f __attribute__((ext_vector_type(8)));

static constexpr int NN  = 100000;     // nodes
static constexpr int EE  = 1200000;    // edges (no self loops)
static constexpr int E2T = EE + NN;    // edges + self loops (GAT)
static constexpr int LM1 = 3;          // L-1 layers
static constexpr int WPB = 5;          // waves per block in GEMM (6250 tiles = 1250*5)

#define DEVFN static __device__ __forceinline__

DEVFN float silu_f(float v)  { return v / (1.f + __expf(-v)); }
DEVFN float lrelu_f(float v) { return v > 0.f ? v : 0.2f * v; }
// Monotonic float -> uint key (for atomicMax-based segment max); key 0 < all real keys.
DEVFN unsigned fkey(float f) {
    unsigned u = __float_as_uint(f);
    return (u & 0x80000000u) ? ~u : (u | 0x80000000u);
}
DEVFN float funkey(unsigned u) {
    return (u & 0x80000000u) ? __uint_as_float(u & 0x7FFFFFFFu) : __uint_as_float(~u);
}

// ------------------------------- small utils -------------------------------

__global__ void k_fill(float* __restrict__ p, float v, int n) {
    int i = blockIdx.x * blockDim.x + threadIdx.x;
    if (i < n) p[i] = v;
}

__global__ void k_deg(const int* __restrict__ dst, float* __restrict__ deg, int e) {
    int i = blockIdx.x * blockDim.x + threadIdx.x;
    if (i < e) atomicAdd(&deg[dst[i]], 1.0f);
}

__global__ void k_invdeg(float* __restrict__ deg, int n) {
    int i = blockIdx.x * blockDim.x + threadIdx.x;
    if (i < n) deg[i] = 1.0f / fmaxf(deg[i], 1.0f);
}

// --------------------------- WMMA fp32 GEMM --------------------------------
// out[nrows x NC] = A[nrows x K] @ W[K x NC] + bias ; optional silu+LayerNorm.
// One wave per 16-row tile; acc layout: VGPR v holds row (v + 8*(lane>=16)),
// col = 16*j + (lane&15). A frag: lane<16 -> K=k,k+1; lane>=16 -> K=k+2,k+3.

template <int K, int NC, bool DO_LN>
__global__ __launch_bounds__(32 * WPB)
void k_gemm(const float* __restrict__ A, const float* __restrict__ W,
            const float* __restrict__ bias, const float* __restrict__ gamma,
            const float* __restrict__ beta, float* __restrict__ out, int ntiles) {
    constexpr int NJ = NC / 16;
    const int wave = threadIdx.x >> 5;
    const int lane = threadIdx.x & 31;
    const int l16  = lane & 15;
    const int half = lane >> 4;                    // 0 or 1
    const int tile = blockIdx.x * WPB + wave;      // uniform per wave
    const bool active = tile < ntiles;

    v8f zero = {};
    v8f acc[NJ];
#pragma unroll
    for (int j = 0; j < NJ; ++j) acc[j] = zero;

    if (active) {
        const float* Arow = A + (size_t)tile * 16 * K + (size_t)l16 * K;
#pragma unroll
        for (int k = 0; k < K; k += 4) {
            const float* ap = Arow + k + 2 * half;
            v2f a;
            a.x = ap[0];
            a.y = ap[1];
#pragma unroll
            for (int j = 0; j < NJ; ++j) {
                const int col = j * 16 + l16;
                v2f b;
                b.x = W[(size_t)(k + 2 * half) * NC + col];
                b.y = W[(size_t)(k + 2 * half + 1) * NC + col];
                acc[j] = __builtin_amdgcn_wmma_f32_16x16x4_f32(
                    false, a, false, b, (short)0, acc[j], false, false);
            }
        }
    }

    if constexpr (DO_LN) {
        static_assert(NC == 64, "LN epilogue assumes 64 columns");
        __shared__ float lds[WPB][16][68];       // pad 68 to dodge bank conflicts
        if (active) {
#pragma unroll
            for (int j = 0; j < NJ; ++j) {
                const int col = j * 16 + l16;
                const float bj = bias[col];
#pragma unroll
                for (int v = 0; v < 8; ++v) {
                    const int r = v + 8 * half;
                    lds[wave][r][col] = silu_f(acc[j][v] + bj);
                }
            }
        }
        __syncthreads();
        if (active) {
            const int r = l16, cb = half * 32;   // lane handles row r, half the cols
            float s = 0.f, sq = 0.f;
#pragma unroll
            for (int c = 0; c < 32; ++c) {
                float vv = lds[wave][r][cb + c];
                s += vv; sq += vv * vv;
            }
            s  += __shfl_xor(s, 16, 32);
            sq += __shfl_xor(sq, 16, 32);
            const float mean = s * (1.f / NC);
            const float var  = sq * (1.f / NC) - mean * mean;
            const float inv  = rsqrtf(var + 1e-5f);
            float* orow = out + ((size_t)tile * 16 + r) * NC;
#pragma unroll
            for (int c = 0; c < 32; ++c) {
                const int col = cb + c;
                const float vv = lds[wave][r][col];
                orow[col] = (vv - mean) * inv * gamma[col] + beta[col];
            }
        }
    } else {
        if (active) {
#pragma unroll
            for (int j = 0; j < NJ; ++j) {
                const int col = j * 16 + l16;
                const float bj = bias[col];
#pragma unroll
                for (int v = 0; v < 8; ++v) {
                    const int r = v + 8 * half;
                    out[((size_t)tile * 16 + r) * NC + col] = acc[j][v] + bj;
                }
            }
        }
        (void)gamma; (void)beta;
    }
}

// ------------------------- encoder mean aggregation ------------------------

__global__ void k_scatter_mean(const int* __restrict__ src, const int* __restrict__ dst,
                               const float* __restrict__ z, float* __restrict__ agg, int e) {
    int t = blockIdx.x * blockDim.x + threadIdx.x;
    if (t >= e * 16) return;
    const int ed = t >> 4, q = t & 15;
    const int s = src[ed], d = dst[ed];
    const float4 v = reinterpret_cast<const float4*>(z + (size_t)s * 64)[q];
    float* ad = agg + (size_t)d * 64 + q * 4;
    atomicAdd(ad + 0, v.x);
    atomicAdd(ad + 1, v.y);
    atomicAdd(ad + 2, v.z);
    atomicAdd(ad + 3, v.w);
}

__global__ void k_mean_res(float* __restrict__ h, const float* __restrict__ agg,
                           const float* __restrict__ invdeg, int n64) {
    int t = blockIdx.x * blockDim.x + threadIdx.x;
    if (t < n64) h[t] += agg[t] * invdeg[t >> 6];
}

// ------------------------------ GATv2 pieces -------------------------------

template <int NC>
__global__ void k_gat_score(const int* __restrict__ src, const int* __restrict__ dst,
                            const float* __restrict__ xl, const float* __restrict__ xr,
                            const float* __restrict__ att, float* __restrict__ esc,
                            unsigned* __restrict__ mu, int e_graph, int ntot) {
    int t = blockIdx.x * blockDim.x + threadIdx.x;
    if (t >= ntot * 16) return;
    const int ed = t >> 4, q = t & 15;
    int s, d;
    if (ed < e_graph) { s = src[ed]; d = dst[ed]; } else { s = d = ed - e_graph; }
    constexpr int CPL = NC / 16;
    float sum = 0.f;
#pragma unroll
    for (int c0 = 0; c0 < CPL; ++c0) {
        const int c = q * CPL + c0;
        sum += att[c] * lrelu_f(xl[(size_t)s * NC + c] + xr[(size_t)d * NC + c]);
    }
#pragma unroll
    for (int off = 8; off >= 1; off >>= 1) sum += __shfl_xor(sum, off, 32);
    if (q == 0) {
        esc[ed] = sum;
        atomicMax(&mu[d], fkey(sum));
    }
}

__global__ void k_gat_w(const int* __restrict__ dst, float* __restrict__ esc,
                        const unsigned* __restrict__ mu, float* __restrict__ denom,
                        int e_graph, int ntot) {
    int ed = blockIdx.x * blockDim.x + threadIdx.x;
    if (ed >= ntot) return;
    const int d = (ed < e_graph) ? dst[ed] : (ed - e_graph);
    const float w = __expf(esc[ed] - funkey(mu[d]));
    esc[ed] = w;
    atomicAdd(&denom[d], w);
}

__global__ void k_gat_coef(const int* __restrict__ dst, float* __restrict__ esc,
                           const float* __restrict__ denom, int e_graph, int ntot) {
    int ed = blockIdx.x * blockDim.x + threadIdx.x;
    if (ed >= ntot) return;
    const int d = (ed < e_graph) ? dst[ed] : (ed - e_graph);
    esc[ed] = esc[ed] / denom[d];
}

template <int NC>
__global__ void k_init_bias(float* __restrict__ z, const float* __restrict__ bias, int total) {
    int t = blockIdx.x * blockDim.x + threadIdx.x;
    if (t < total) z[t] = bias[t & (NC - 1)];
}

template <int NC>
__global__ void k_gat_scatter(const int* __restrict__ src, const int* __restrict__ dst,
                              const float* __restrict__ xl, const float* __restrict__ coef,
                              float* __restrict__ z, int e_graph, int ntot) {
    constexpr int TPE = NC / 4;
    int t = blockIdx.x * blockDim.x + threadIdx.x;
    if (t >= ntot * TPE) return;
    const int ed = t / TPE, q = t - ed * TPE;
    int s, d;
    if (ed < e_graph) { s = src[ed]; d = dst[ed]; } else { s = d = ed - e_graph; }
    const float a = coef[ed];
    const float4 v = reinterpret_cast<const float4*>(xl + (size_t)s * NC)[q];
    float* zp = z + (size_t)d * NC + q * 4;
    atomicAdd(zp + 0, a * v.x);
    atomicAdd(zp + 1, a * v.y);
    atomicAdd(zp + 2, a * v.z);
    atomicAdd(zp + 3, a * v.w);
}

// out = LN(silu(z))*g + be (+ optional residual hin), 16 lanes per row.
template <int NC, bool RES>
__global__ void k_gat_post(const float* __restrict__ z, const float* __restrict__ g,
                           const float* __restrict__ be, const float* __restrict__ hin,
                           float* __restrict__ out, int n) {
    int t = blockIdx.x * blockDim.x + threadIdx.x;
    if (t >= n * 16) return;
    const int row = t >> 4, q = t & 15;
    constexpr int CPL = NC / 16;
    float vals[CPL];
    float s = 0.f, sq = 0.f;
#pragma unroll
    for (int c0 = 0; c0 < CPL; ++c0) {
        const float vv = silu_f(z[(size_t)row * NC + q * CPL + c0]);
        vals[c0] = vv;
        s += vv; sq += vv * vv;
    }
#pragma unroll
    for (int off = 8; off >= 1; off >>= 1) {
        s  += __shfl_xor(s, off, 32);
        sq += __shfl_xor(sq, off, 32);
    }
    const float mean = s * (1.f / NC);
    const float var  = sq * (1.f / NC) - mean * mean;
    const float inv  = rsqrtf(var + 1e-5f);
#pragma unroll
    for (int c0 = 0; c0 < CPL; ++c0) {
        const int col = q * CPL + c0;
        float o = (vals[c0] - mean) * inv * g[col] + be[col];
        if (RES) o += hin[(size_t)row * NC + col];
        out[(size_t)row * NC + col] = o;
    }
}

// -------------------------------- launcher ---------------------------------

extern "C" void kernel_launch(void* const* d_in, const int* in_sizes, int n_in,
                              void* d_out, int out_size, void* d_ws, size_t ws_size,
                              hipStream_t stream) {
    (void)in_sizes; (void)n_in; (void)out_size; (void)ws_size;

    const float* x        = (const float*)d_in[0];
    const int*   ei       = (const int*)  d_in[1];
    const float* W_emb    = (const float*)d_in[2];
    const float* b_emb    = (const float*)d_in[3];
    const float* g_emb    = (const float*)d_in[4];
    const float* be_emb   = (const float*)d_in[5];
    const float* enc_W    = (const float*)d_in[6];
    const float* enc_b    = (const float*)d_in[7];
    const float* enc_g    = (const float*)d_in[8];
    const float* enc_be   = (const float*)d_in[9];
    const float* dec_Wl   = (const float*)d_in[10];
    const float* dec_bl   = (const float*)d_in[11];
    const float* dec_Wr   = (const float*)d_in[12];
    const float* dec_br   = (const float*)d_in[13];
    const float* dec_att  = (const float*)d_in[14];
    const float* dec_bias = (const float*)d_in[15];
    const float* dec_g    = (const float*)d_in[16];
    const float* dec_be   = (const float*)d_in[17];
    const float* fin_Wl   = (const float*)d_in[18];
    const float* fin_bl   = (const float*)d_in[19];
    const float* fin_Wr   = (const float*)d_in[20];
    const float* fin_br   = (const float*)d_in[21];
    const float* fin_att  = (const float*)d_in[22];
    const float* fin_bias = (const float*)d_in[23];
    const float* fin_g    = (const float*)d_in[24];
    const float* fin_be   = (const float*)d_in[25];

    const int* src = ei;
    const int* dst = ei + EE;
    float* out = (float*)d_out;

    // workspace layout (all 16B aligned: every size is a multiple of 4 floats)
    float* w       = (float*)d_ws;
    float*    invdeg = w;               w += NN;
    float*    denom  = w;               w += NN;
    unsigned* mu     = (unsigned*)w;    w += NN;
    float*    h      = w;               w += (size_t)NN * 64;
    float*    zb     = w;               w += (size_t)NN * 64;
    float*    xl     = w;               w += (size_t)NN * 64;   // also encoder agg
    float*    xr     = w;               w += (size_t)NN * 64;
    float*    esc    = w;               w += E2T;

    const int TB = 256;
    auto nb = [](long long t) { return (unsigned)((t + 255) / 256); };
    const int NT = NN / 16;              // 6250 row-tiles
    const unsigned GG = (unsigned)((NT + WPB - 1) / WPB);   // 1250 blocks, exact fit
    const dim3 gthr(32 * WPB);

    // ---- in-degree -> 1/max(deg,1)
    k_fill<<<nb(NN), TB, 0, stream>>>(invdeg, 0.f, NN);
    k_deg<<<nb(EE), TB, 0, stream>>>(dst, invdeg, EE);
    k_invdeg<<<nb(NN), TB, 0, stream>>>(invdeg, NN);

    // ---- initial embedding: h = LN(silu(x @ W_emb + b))
    k_gemm<32, 64, true><<<GG, gthr, 0, stream>>>(x, W_emb, b_emb, g_emb, be_emb, h, NT);

    // ---- encoder layers: h = segmean(LN(silu(h@W+b))[src] -> dst) + h
    for (int i = 0; i < LM1; ++i) {
        k_gemm<64, 64, true><<<GG, gthr, 0, stream>>>(
            h, enc_W + (size_t)i * 64 * 64, enc_b + i * 64, enc_g + i * 64, enc_be + i * 64, zb, NT);
        k_fill<<<nb((long long)NN * 64), TB, 0, stream>>>(xl, 0.f, NN * 64);
        k_scatter_mean<<<nb((long long)EE * 16), TB, 0, stream>>>(src, dst, zb, xl, EE);
        k_mean_res<<<nb((long long)NN * 64), TB, 0, stream>>>(h, xl, invdeg, NN * 64);
    }

    // ---- decoder GATv2 layers (with self loops), residual
    for (int i = 0; i < LM1; ++i) {
        k_gemm<64, 64, false><<<GG, gthr, 0, stream>>>(
            h, dec_Wl + (size_t)i * 64 * 64, dec_bl + i * 64, nullptr, nullptr, xl, NT);
        k_gemm<64, 64, false><<<GG, gthr, 0, stream>>>(
            h, dec_Wr + (size_t)i * 64 * 64, dec_br + i * 64, nullptr, nullptr, xr, NT);
        k_fill<<<nb(NN), TB, 0, stream>>>((float*)mu, 0.f, NN);
        k_fill<<<nb(NN), TB, 0, stream>>>(denom, 0.f, NN);
        k_gat_score<64><<<nb((long long)E2T * 16), TB, 0, stream>>>(
            src, dst, xl, xr, dec_att + i * 64, esc, mu, EE, E2T);
        k_gat_w<<<nb(E2T), TB, 0, stream>>>(dst, esc, mu, denom, EE, E2T);
        k_gat_coef<<<nb(E2T), TB, 0, stream>>>(dst, esc, denom, EE, E2T);
        k_init_bias<64><<<nb((long long)NN * 64), TB, 0, stream>>>(zb, dec_bias + i * 64, NN * 64);
        k_gat_scatter<64><<<nb((long long)E2T * 16), TB, 0, stream>>>(src, dst, xl, esc, zb, EE, E2T);
        k_gat_post<64, true><<<nb((long long)NN * 16), TB, 0, stream>>>(
            zb, dec_g + i * 64, dec_be + i * 64, h, h, NN);
    }

    // ---- final GATv2 (H -> D_IN=32), then LN(silu(z)) -> out
    k_gemm<64, 32, false><<<GG, gthr, 0, stream>>>(h, fin_Wl, fin_bl, nullptr, nullptr, xl, NT);
    k_gemm<64, 32, false><<<GG, gthr, 0, stream>>>(h, fin_Wr, fin_br, nullptr, nullptr, xr, NT);
    k_fill<<<nb(NN), TB, 0, stream>>>((float*)mu, 0.f, NN);
    k_fill<<<nb(NN), TB, 0, stream>>>(denom, 0.f, NN);
    k_gat_score<32><<<nb((long long)E2T * 16), TB, 0, stream>>>(
        src, dst, xl, xr, fin_att, esc, mu, EE, E2T);
    k_gat_w<<<nb(E2T), TB, 0, stream>>>(dst, esc, mu, denom, EE, E2T);
    k_gat_coef<<<nb(E2T), TB, 0, stream>>>(dst, esc, denom, EE, E2T);
    k_init_bias<32><<<nb((long long)NN * 32), TB, 0, stream>>>(zb, fin_bias, NN * 32);
    k_gat_scatter<32><<<nb((long long)E2T * 8), TB, 0, stream>>>(src, dst, xl, esc, zb, EE, E2T);
    k_gat_post<32, false><<<nb((long long)NN * 16), TB, 0, stream>>>(
        zb, fin_g, fin_be, nullptr, out, NN);
}